// KPlaneRBFField_15917148799750
// MI455X (gfx1250) — compile-verified
//
#include <hip/hip_runtime.h>
#include <cstdint>

#define NSIDE 64
#define WAVES_PER_BLOCK 8
#define PTS_PER_WAVE 4

typedef unsigned int v4u __attribute__((ext_vector_type(4)));
typedef int v8i __attribute__((ext_vector_type(8)));
typedef int v4i __attribute__((ext_vector_type(4)));

struct Ax { int i; float w; };

// align_corners=true grid coordinate for one axis: index + fractional weight.
__device__ __forceinline__ Ax axis_coord(float u, int R) {
    float g = (u + 1.0f) * 0.5f * (float)(R - 1);
    float f = fminf(fmaxf(floorf(g), 0.0f), (float)(R - 2));
    Ax a; a.i = (int)f; a.w = g - f; return a;
}

// Bilinear sample with wave-uniform base pushed to SGPRs (saddr form) and the
// per-lane channel offset (lane*R*R) hoisted by the caller.
__device__ __forceinline__ float bsample(const float* __restrict__ p, int R, int laneOff,
                                         Ax ax, Ax ay) {
    int so = __builtin_amdgcn_readfirstlane(ay.i * R + ax.i);
    const float* b = p + so;
    float v00 = b[laneOff];
    float v01 = b[laneOff + 1];
    float v10 = b[laneOff + R];
    float v11 = b[laneOff + R + 1];
    float wx = ax.w, wy = ay.w;
    float omx = 1.0f - wx, omy = 1.0f - wy;
    return v00 * (omx * omy) + v01 * (wx * omy) + v10 * (omx * wy) + v11 * (wx * wy);
}

__global__ __launch_bounds__(256) void kplane_rbf_kernel(
    const float* __restrict__ pts, const float* __restrict__ aabb,
    const float* __restrict__ p00, const float* __restrict__ p01, const float* __restrict__ p02,
    const float* __restrict__ p10, const float* __restrict__ p11, const float* __restrict__ p12,
    const float* __restrict__ p20, const float* __restrict__ p21, const float* __restrict__ p22,
    const float* __restrict__ lc0, const float* __restrict__ ks,
    const float* __restrict__ kpb, float* __restrict__ out, int N)
{
    __shared__ __align__(16) float sbias[128];
    // Per-wave slab: 8 gathered code rows of 128B each, filled by the TDM.
    __shared__ __align__(128) float scode[WAVES_PER_BLOCK][8][32];

    const int lane = threadIdx.x & 31;
    const int wv   = __builtin_amdgcn_readfirstlane((int)(threadIdx.x >> 5));

    // Stage the 128-float bias into LDS: one gfx1250 async B128 per lane of wave 0
    // (GVS mode: saddr + vgpr byte offset), tracked by ASYNCcnt.
    if (wv == 0) {
        unsigned lds_off = (unsigned)(uintptr_t)(&sbias[0]) + (unsigned)(lane * 16);
        unsigned goff    = (unsigned)(lane * 16);
        asm volatile(
            "global_load_async_to_lds_b128 %0, %1, %2\n\t"
            "s_wait_asynccnt 0x0"
            :
            : "v"(lds_off), "v"(goff), "s"(kpb)
            : "memory");
    }
    __syncthreads();

    const float bias0 = sbias[lane];
    const float bias1 = sbias[32 + lane];
    const float bias2 = sbias[64 + lane];
    const float bias3 = sbias[96 + lane];

    // Per-lane channel offsets (elements), loop-invariant: lane * R * R.
    const int lo0 = lane * (128 * 128);
    const int lo1 = lane * (256 * 256);
    const int lo2 = lane * (512 * 512);

    // aabb normalization constants (scalar loads).
    const float a0x = aabb[0], a0y = aabb[1], a0z = aabb[2];
    const float a1x = aabb[3], a1y = aabb[4], a1z = aabb[5];
    const float scx = 2.0f / (a1x - a0x);
    const float scy = 2.0f / (a1y - a0y);
    const float scz = 2.0f / (a1z - a0z);

    const unsigned lds_slab = (unsigned)(uintptr_t)(&scode[wv][0][0]);
    const uint64_t ga = (uint64_t)(uintptr_t)lc0;

    int n = (blockIdx.x * WAVES_PER_BLOCK + wv) * PTS_PER_WAVE;
    for (int t = 0; t < PTS_PER_WAVE; ++t, ++n) {
        if (n >= N) break;

        // ---- normalize point (n is SGPR-uniform -> pts come in via s_load) ----
        float xn = (pts[3 * n + 0] - a0x) * scx - 1.0f;
        float yn = (pts[3 * n + 1] - a0y) * scy - 1.0f;
        float zn = (pts[3 * n + 2] - a0z) * scz - 1.0f;

        // ---- RBF cell (IEEE div kept: feeds a discontinuous floor) ----
        const float interval = 2.0f / 63.0f;
        float cxf = fminf(fmaxf(floorf((xn + 1.0f) / interval), 0.0f), 62.0f);
        float cyf = fminf(fmaxf(floorf((yn + 1.0f) / interval), 0.0f), 62.0f);
        float czf = fminf(fmaxf(floorf((zn + 1.0f) / interval), 0.0f), 62.0f);
        int cx = (int)cxf, cy = (int)cyf, cz = (int)czf;

        // Corner ids (OFFS order; strictly increasing), wave-uniform in SGPRs.
        int base_id = (cx * NSIDE + cy) * NSIDE + cz;
        int ids[8];
#pragma unroll
        for (int c = 0; c < 8; ++c) {
            int i = (c >> 2) & 1, j = (c >> 1) & 1, k = c & 1;
            ids[c] = __builtin_amdgcn_readfirstlane(
                base_id + i * (NSIDE * NSIDE) + j * NSIDE + k);
        }

        // ---- TDM gather-mode load: 8 x 128B code rows -> LDS (TENSORcnt) ----
        // Issued now so the DMA overlaps the entire plane-sampling phase below.
        // Inline asm (portable across both toolchains' builtin arities):
        // VADDR0=group0 (4 SGPRs), VADDR1=group1 (8 SGPRs), VADDR2/3=groups 2/3.
        v4u g0; v8i g1; v4i g2, g3;
        g0[0] = 1u | (1u << 30) | (1u << 31);            // count=1, 32b idx, gather_mode
        g0[1] = lds_slab;                                // lds_addr
        g0[2] = (unsigned)(ga & 0xffffffffu);            // global_addr[31:0]
        g0[3] = (unsigned)((ga >> 32) & 0x01ffffffu)     // global_addr[56:32]
              | (2u << 30);                              // type = 2 (image)
        g1[0] = (int)(2u << 16);                         // wg_mask=0, data_size=4B
        g1[1] = (int)(32u << 16);                        // tensor_dim0 = 32
        g1[2] = 0;                                       // tensor_dim1 lo16 = 0
        g1[3] = (int)(4u | (32u << 16));                 // tensor_dim1=64^3, tile_dim0=32
        g1[4] = 8;                                       // tile_dim1 = 8 valid indices
        g1[5] = 32;                                      // tensor_dim0_stride = 32
        g1[6] = 0;
        g1[7] = 0;
        g2[0] = ids[0]; g2[1] = ids[1]; g2[2] = ids[2]; g2[3] = ids[3];
        g3[0] = ids[4]; g3[1] = ids[5]; g3[2] = ids[6]; g3[3] = ids[7];
        asm volatile(
            "tensor_load_to_lds %0, %1, %2, %3"
            :
            : "s"(g0), "s"(g1), "s"(g2), "s"(g3)
            : "memory");

        // ---- phi weights (ks via SMEM; smooth fn -> 1-ULP v_rcp ok) ----
        float dx0 = xn - (-1.0f + cxf * interval);
        float dx1 = xn - (-1.0f + (cxf + 1.0f) * interval);
        float dy0 = yn - (-1.0f + cyf * interval);
        float dy1 = yn - (-1.0f + (cyf + 1.0f) * interval);
        float dz0 = zn - (-1.0f + czf * interval);
        float dz1 = zn - (-1.0f + (czf + 1.0f) * interval);
        float sx0 = dx0 * dx0, sx1 = dx1 * dx1;
        float sy0 = dy0 * dy0, sy1 = dy1 * dy1;
        float sz0 = dz0 * dz0, sz1 = dz1 * dz1;

        float phi[8];
        float psum = 0.0f;
#pragma unroll
        for (int c = 0; c < 8; ++c) {
            int i = (c >> 2) & 1, j = (c >> 1) & 1, k = c & 1;
            float d2 = (i ? sx1 : sx0) + (j ? sy1 : sy0) + (k ? sz1 : sz0);
            float s  = ks[ids[c]];
            float ph = __builtin_amdgcn_rcpf(fmaf(d2, s * s, 1.0f));
            phi[c] = ph;
            psum  += ph;
        }

        // ---- K-plane bilinear products (lane = channel; TDM runs underneath) ----
        Ax x0a = axis_coord(xn, 128), y0a = axis_coord(yn, 128), z0a = axis_coord(zn, 128);
        float f0 = bsample(p00, 128, lo0, x0a, y0a)
                 * bsample(p01, 128, lo0, x0a, z0a)
                 * bsample(p02, 128, lo0, y0a, z0a);
        Ax x1a = axis_coord(xn, 256), y1a = axis_coord(yn, 256), z1a = axis_coord(zn, 256);
        float f1 = bsample(p10, 256, lo1, x1a, y1a)
                 * bsample(p11, 256, lo1, x1a, z1a)
                 * bsample(p12, 256, lo1, y1a, z1a);
        Ax x2a = axis_coord(xn, 512), y2a = axis_coord(yn, 512), z2a = axis_coord(zn, 512);
        float f2 = bsample(p20, 512, lo2, x2a, y2a)
                 * bsample(p21, 512, lo2, x2a, z2a)
                 * bsample(p22, 512, lo2, y2a, z2a);

        // ---- RBF weighted sum ----
        // The TDM's LDS write is invisible to alias analysis (the address
        // escaped as an integer in the descriptor), so read the rows back
        // with explicit ds_loads fenced by the tensor/DS counters.
        float rbfu;
        {
            unsigned vaddr = lds_slab + (unsigned)(lane * 4);
            float c0, c1, c2, c3, c4, c5, c6, c7;
            asm volatile(
                "s_wait_tensorcnt 0x0\n\t"
                "ds_load_b32 %0, %8 offset:0\n\t"
                "ds_load_b32 %1, %8 offset:128\n\t"
                "ds_load_b32 %2, %8 offset:256\n\t"
                "ds_load_b32 %3, %8 offset:384\n\t"
                "ds_load_b32 %4, %8 offset:512\n\t"
                "ds_load_b32 %5, %8 offset:640\n\t"
                "ds_load_b32 %6, %8 offset:768\n\t"
                "ds_load_b32 %7, %8 offset:896\n\t"
                "s_wait_dscnt 0x0"
                : "=v"(c0), "=v"(c1), "=v"(c2), "=v"(c3),
                  "=v"(c4), "=v"(c5), "=v"(c6), "=v"(c7)
                : "v"(vaddr)
                : "memory");
            rbfu = phi[0] * c0;
            rbfu = fmaf(phi[1], c1, rbfu);
            rbfu = fmaf(phi[2], c2, rbfu);
            rbfu = fmaf(phi[3], c3, rbfu);
            rbfu = fmaf(phi[4], c4, rbfu);
            rbfu = fmaf(phi[5], c5, rbfu);
            rbfu = fmaf(phi[6], c6, rbfu);
            rbfu = fmaf(phi[7], c7, rbfu);
        }
        float rbf = rbfu * __builtin_amdgcn_rcpf(psum + 1e-8f);

        // ---- bias + non-temporal streamed stores (keep L2 for the gather tables) ----
        size_t base = (size_t)n * 128;
        __builtin_nontemporal_store(f0  + bias0, &out[base + 0  + lane]);
        __builtin_nontemporal_store(f1  + bias1, &out[base + 32 + lane]);
        __builtin_nontemporal_store(f2  + bias2, &out[base + 64 + lane]);
        __builtin_nontemporal_store(rbf + bias3, &out[base + 96 + lane]);
    }
}

extern "C" void kernel_launch(void* const* d_in, const int* in_sizes, int n_in,
                              void* d_out, int out_size, void* d_ws, size_t ws_size,
                              hipStream_t stream) {
    const float* pts  = (const float*)d_in[0];
    const float* aabb = (const float*)d_in[1];
    const float* p00  = (const float*)d_in[2];
    const float* p01  = (const float*)d_in[3];
    const float* p02  = (const float*)d_in[4];
    const float* p10  = (const float*)d_in[5];
    const float* p11  = (const float*)d_in[6];
    const float* p12  = (const float*)d_in[7];
    const float* p20  = (const float*)d_in[8];
    const float* p21  = (const float*)d_in[9];
    const float* p22  = (const float*)d_in[10];
    const float* lc0  = (const float*)d_in[11];
    const float* ks   = (const float*)d_in[12];
    const float* kpb  = (const float*)d_in[13];
    float* out = (float*)d_out;

    int N = in_sizes[0] / 3;
    int ptsPerBlock = WAVES_PER_BLOCK * PTS_PER_WAVE;          // 32
    int blocks = (N + ptsPerBlock - 1) / ptsPerBlock;
    hipLaunchKernelGGL(kplane_rbf_kernel, dim3(blocks), dim3(256), 0, stream,
                       pts, aabb, p00, p01, p02, p10, p11, p12, p20, p21, p22,
                       lc0, ks, kpb, out, N);
}